// UpsampleBlock_7842610283218
// MI455X (gfx1250) — compile-verified
//
#include <hip/hip_runtime.h>

typedef __attribute__((ext_vector_type(2))) float v2f;
typedef __attribute__((ext_vector_type(8))) float v8f;

#define N1 1024   // reference points per batch
#define N2 4096   // query points per batch
#define C1T 259   // 3 + 256
#define F1 256
#define C2T 131   // 3 + 128
#define F2 128
#define OC 387    // 3 + 256 + 128
#define NB 8
#define KEPS 1e-7f

__global__ __launch_bounds__(256) void upsample_knn3_wmma(
    const float* __restrict__ in1,   // (8,1024,259)
    const float* __restrict__ in2,   // (8,4096,131)
    float* __restrict__ out)         // (8,4096,387) then (8,4096,3)
{
    __shared__ float4 sref[N1];      // (x,y,z,|p|^2) : 16 KB

    const int b    = blockIdx.y;
    const int tid  = threadIdx.x;
    const int lane = tid & 31;
    const int wave = tid >> 5;
    const int hh   = (lane >> 4);    // 0: lanes 0-15, 1: lanes 16-31
    const int ln   = lane & 15;

    const float* in1b = in1 + (size_t)b * N1 * C1T;
    const float* in2b = in2 + (size_t)b * N2 * C2T;

    // Stage xyz1 + squared norm into LDS (shared by all 8 waves)
    for (int i = tid; i < N1; i += 256) {
        const float* p = in1b + i * C1T;
        float x = p[0], y = p[1], z = p[2];
        sref[i] = make_float4(x, y, z, x * x + y * y + z * z);
    }
    __syncthreads();

    // This wave owns 16 query points
    const int qb = blockIdx.x * 128 + wave * 16;
    const int q  = qb + ln;
    const float* qp = in2b + q * C2T;
    float qx = qp[0], qy = qp[1], qz = qp[2];
    float q2 = qx * qx + qy * qy + qz * qz;

    // B matrix (4 x 16): rows (-2x, -2y, -2z, 1) per query column.
    // VGPR0: lanes0-15 = B[0][n], lanes16-31 = B[2][n]
    // VGPR1: lanes0-15 = B[1][n], lanes16-31 = B[3][n]
    v2f bm;
    bm.x = hh ? (-2.0f * qz) : (-2.0f * qx);
    bm.y = hh ? 1.0f         : (-2.0f * qy);

    float bd0 = 3.4e38f, bd1 = 3.4e38f, bd2 = 3.4e38f;
    int   bi0 = 0, bi1 = 0, bi2 = 0;

    for (int tb = 0; tb < N1; tb += 16) {
        // A matrix (16 refs x 4): (x, y, z, |r|^2)
        // VGPR0: lanes0-15 = A[m][0], lanes16-31 = A[m][2]
        // VGPR1: lanes0-15 = A[m][1], lanes16-31 = A[m][3]
        float4 rr = sref[tb + ln];
        v2f am;
        am.x = hh ? rr.z : rr.x;
        am.y = hh ? rr.w : rr.y;

        // C = |q|^2 splat (column index of C == lane&15)
        v8f c;
        c[0] = q2; c[1] = q2; c[2] = q2; c[3] = q2;
        c[4] = q2; c[5] = q2; c[6] = q2; c[7] = q2;

        // D[m][n] = |q_n|^2 - 2 q_n . r_m + |r_m|^2
        v8f d = __builtin_amdgcn_wmma_f32_16x16x4_f32(
            false, am, false, bm, 0, c, false, false);

        // Each lane holds 8 candidates for its query:
        // vgpr r -> ref row (tb + r + 8*hh)
        int ibase = tb + hh * 8;
#pragma unroll
        for (int r8 = 0; r8 < 8; ++r8) {
            float dv = d[r8];
            int   iv = ibase + r8;
            bool lt2 = dv < bd2;
            bool lt1 = dv < bd1;
            bool lt0 = dv < bd0;
            bd2 = lt1 ? bd1 : (lt2 ? dv : bd2);
            bi2 = lt1 ? bi1 : (lt2 ? iv : bi2);
            bd1 = lt0 ? bd0 : (lt1 ? dv : bd1);
            bi1 = lt0 ? bi0 : (lt1 ? iv : bi1);
            bd0 = lt0 ? dv : bd0;
            bi0 = lt0 ? iv : bi0;
        }
    }

    // Merge lane pair (l, l+16): each covered half the ref rows
    {
        float od[3]; int oi[3];
        od[0] = __shfl_xor(bd0, 16, 32); oi[0] = __shfl_xor(bi0, 16, 32);
        od[1] = __shfl_xor(bd1, 16, 32); oi[1] = __shfl_xor(bi1, 16, 32);
        od[2] = __shfl_xor(bd2, 16, 32); oi[2] = __shfl_xor(bi2, 16, 32);
#pragma unroll
        for (int k = 0; k < 3; ++k) {
            float dv = od[k];
            int   iv = oi[k];
            bool lt2 = dv < bd2;
            bool lt1 = dv < bd1;
            bool lt0 = dv < bd0;
            bd2 = lt1 ? bd1 : (lt2 ? dv : bd2);
            bi2 = lt1 ? bi1 : (lt2 ? iv : bi2);
            bd1 = lt0 ? bd0 : (lt1 ? dv : bd1);
            bi1 = lt0 ? bi0 : (lt1 ? iv : bi1);
            bd0 = lt0 ? dv : bd0;
            bi0 = lt0 ? iv : bi0;
        }
    }

    // Inverse-distance weights
    float t0 = fmaxf(bd0, KEPS), t1 = fmaxf(bd1, KEPS), t2 = fmaxf(bd2, KEPS);
    float v0 = 1.0f / t0, v1 = 1.0f / t1, vv2 = 1.0f / t2;
    float wsum = v0 + v1 + vv2;
    float w0 = v0 / wsum, w1 = v1 / wsum, w2 = vv2 / wsum;

    // Cooperative, coalesced interpolation + copy for this wave's 16 queries
    float* out1p = out;
    float* out2p = out + (size_t)NB * N2 * OC;

#pragma unroll 1
    for (int j = 0; j < 16; ++j) {
        int qq = qb + j;
        int gi0 = __shfl(bi0, j, 32);
        int gi1 = __shfl(bi1, j, 32);
        int gi2 = __shfl(bi2, j, 32);
        float gw0 = __shfl(w0, j, 32);
        float gw1 = __shfl(w1, j, 32);
        float gw2 = __shfl(w2, j, 32);

        const float* f0 = in1b + (size_t)gi0 * C1T + 3;
        const float* f1 = in1b + (size_t)gi1 * C1T + 3;
        const float* f2 = in1b + (size_t)gi2 * C1T + 3;
        const float* qrow = in2b + (size_t)qq * C2T;
        float* orow = out1p + ((size_t)b * N2 + qq) * OC;

#pragma unroll
        for (int c = lane; c < F1; c += 32)
            orow[3 + c] = gw0 * f0[c] + gw1 * f1[c] + gw2 * f2[c];
#pragma unroll
        for (int c = lane; c < F2; c += 32)
            orow[3 + F1 + c] = qrow[3 + c];
        if (lane < 3) {
            orow[lane] = qrow[lane];
            out2p[((size_t)b * N2 + qq) * 3 + lane] = qrow[lane];
        }
    }
}

extern "C" void kernel_launch(void* const* d_in, const int* in_sizes, int n_in,
                              void* d_out, int out_size, void* d_ws, size_t ws_size,
                              hipStream_t stream) {
    const float* in1 = (const float*)d_in[0];   // (8,1024,259) f32
    const float* in2 = (const float*)d_in[1];   // (8,4096,131) f32
    float* out = (float*)d_out;                 // 8*4096*387 + 8*4096*3 f32

    dim3 grid(N2 / 128, NB);   // 32 query-chunks x 8 batches
    upsample_knn3_wmma<<<grid, 256, 0, stream>>>(in1, in2, out);
}